// Potts_Energy_Fn_19688130085114
// MI455X (gfx1250) — compile-verified
//
#include <hip/hip_runtime.h>

typedef float v2f __attribute__((ext_vector_type(2)));
typedef float v8f __attribute__((ext_vector_type(8)));

#define NN   1024      // N
#define BB   128       // batch
#define LDSW 1028      // padded LDS row stride (floats): +4 -> conflict-free b64 reads
#define NBLK (NN / 16) // 64 i-tiles

// ds_swizzle group-of-32: offset = {0, xor[14:10], or[9:5], and[4:0]}
template <int PAT>
__device__ __forceinline__ float swz_xor(float x) {
    return __int_as_float(__builtin_amdgcn_ds_swizzle(__float_as_int(x), PAT));
}

// P = A_stack(256x1024) * triu(W)^T, fused with the row-dot epilogue.
// A_stack rows 0..127 = 1 - vector[b,:], rows 128..255 = vector[b,:].
__global__ __launch_bounds__(512) void potts_wmma(const float* __restrict__ vec,
                                                  const float* __restrict__ W,
                                                  float* __restrict__ partial)
{
    __shared__ float lds_w[16 * LDSW];   // 16 rows of triu(W), masked, padded
    __shared__ float lds_e[BB];          // per-workgroup energy accumulator

    const int tid = threadIdx.x;
    const int i0  = blockIdx.x * 16;     // this tile covers W rows [i0, i0+16)
    const int j0  = i0 & ~31;            // K start rounded to 32 (extra cols mask to 0)

    if (tid < BB) lds_e[tid] = 0.0f;

    // ---- stage triu(W) tile rows [i0,i0+16) x cols [j0,1024) into LDS ----
    const int ncol4 = (NN - j0) >> 2;
    for (int idx = tid; idx < 16 * ncol4; idx += 512) {
        const int r    = idx / ncol4;
        const int c4   = idx - r * ncol4;
        const int col  = j0 + (c4 << 2);
        const int irow = i0 + r;
        const float4 wv = *(const float4*)(W + irow * NN + col);
        float4 m;
        m.x = (col + 0 >= irow) ? wv.x : 0.0f;
        m.y = (col + 1 >= irow) ? wv.y : 0.0f;
        m.z = (col + 2 >= irow) ? wv.z : 0.0f;
        m.w = (col + 3 >= irow) ? wv.w : 0.0f;
        *(float4*)&lds_w[r * LDSW + col] = m;
    }
    __syncthreads();

    const int  lane   = tid & 31;
    const int  wave   = tid >> 5;            // 0..15; waves 0-7 = U half, 8-15 = V half
    const int  lane15 = lane & 15;
    const int  khi    = (lane >= 16) ? 2 : 0; // K offset for upper lane half (f32 A/B layout)
    const bool isU    = (wave < 8);

    // a = isU ? 1-v : v  ==  fma(sgn, v, off) with wave-uniform sgn/off -> v_pk_fma_f32
    const float sgn = isU ? -1.0f : 1.0f;
    const float off = isU ?  1.0f : 0.0f;

    // A fragment source row (vector row), per f32 16x4 A layout: M = lane&15
    const float* aRow = vec + ((((wave << 4) + lane15) & (BB - 1)) * NN);

    v8f acc[4] = {v8f{}, v8f{}, v8f{}, v8f{}};

    // ---- K loop, software-pipelined: prefetch next 8 A-fragments (one load
    //      clause into distinct regs) while 8 WMMAs consume the current ones ----
    v2f a_f[8];
#pragma unroll
    for (int u = 0; u < 8; ++u)
        a_f[u] = *(const v2f*)(aRow + j0 + 4 * u + khi);

    for (int j = j0; j < NN; j += 32) {
        const int jn = (j + 32 < NN) ? (j + 32) : j;  // clamped prefetch (last iter: dummy)
        v2f a_n[8];
#pragma unroll
        for (int u = 0; u < 8; ++u)
            a_n[u] = *(const v2f*)(aRow + jn + 4 * u + khi);

#pragma unroll
        for (int u = 0; u < 8; ++u) {
            v2f av = a_f[u];
            av.x = __builtin_fmaf(sgn, av.x, off);
            av.y = __builtin_fmaf(sgn, av.y, off);
            const v2f bv = *(const v2f*)&lds_w[lane15 * LDSW + j + 4 * u + khi];
            acc[u & 3] = __builtin_amdgcn_wmma_f32_16x16x4_f32(
                            false, av, false, bv, (short)0, acc[u & 3], false, false);
        }
#pragma unroll
        for (int u = 0; u < 8; ++u) a_f[u] = a_n[u];
    }
    const v8f c = (acc[0] + acc[1]) + (acc[2] + acc[3]);

    // ---- epilogue: energy += A[m, i0+n] * P[m, n], reduced per batch row ----
    const int colN = i0 + lane15;            // N index held by this lane (C/D layout)
#pragma unroll
    for (int r = 0; r < 8; ++r) {
        const int m_local = r + ((lane >= 16) ? 8 : 0);    // M index in C/D VGPR r
        const int mrow    = ((wave << 4) + m_local) & (BB - 1);
        float av = vec[mrow * NN + colN];
        if (isU) av = 1.0f - av;
        float p = av * c[r];
        // XOR-swizzle reduction across each 16-lane half (single ds_swizzle each)
        p += swz_xor<0x041F>(p);  // xor 1
        p += swz_xor<0x081F>(p);  // xor 2
        p += swz_xor<0x101F>(p);  // xor 4
        p += swz_xor<0x201F>(p);  // xor 8
        if (lane15 == 0) {
            // exactly two commutative adds per slot (wave w and w+8) -> deterministic
            atomicAdd(&lds_e[((wave & 7) << 4) + m_local], p);
        }
    }
    __syncthreads();
    if (tid < BB) partial[blockIdx.x * BB + tid] = lds_e[tid];
}

__global__ void potts_reduce(const float* __restrict__ partial,
                             float* __restrict__ out)
{
    const int b = threadIdx.x;
    float s = 0.0f;
#pragma unroll 8
    for (int g = 0; g < NBLK; ++g) s += partial[g * BB + b];
    out[b] = s;
}

extern "C" void kernel_launch(void* const* d_in, const int* in_sizes, int n_in,
                              void* d_out, int out_size, void* d_ws, size_t ws_size,
                              hipStream_t stream)
{
    const float* vec = (const float*)d_in[0];   // (128, 1024) fp32
    const float* W   = (const float*)d_in[1];   // (1024, 1024) fp32
    float* out       = (float*)d_out;           // (128,) fp32
    float* partial   = (float*)d_ws;            // 64*128 floats = 32 KB scratch

    potts_wmma<<<NBLK, 512, 0, stream>>>(vec, W, partial);
    potts_reduce<<<1, BB, 0, stream>>>(partial, out);
}